// EKF_63067299774705
// MI455X (gfx1250) — compile-verified
//
#include <hip/hip_runtime.h>
#include <hip/hip_bf16.h>

typedef __attribute__((ext_vector_type(2))) float v2f;
typedef __attribute__((ext_vector_type(8))) float v8f;

#define DEV static __device__ __forceinline__

DEV float rdl(float v, int lane) {
  // literal-lane broadcast: v_readlane_b32 (no LDS, no counters)
  return __int_as_float(__builtin_amdgcn_readlane(__float_as_int(v), lane));
}

// Per-lane 0/1 weights encoding the WMMA operand layouts (loop-invariant).
// wl_j: lane is in lanes 0-15 and (lane&15)==j ; wh_j: lanes 16-31 and ==j
struct LaneW {
  float l0, l1, l2, l3;
  float h0, h1, h2, h3;
};

// A-tile (16x4 f32) of row-major 3x3 F: A(m,k)=F[m*3+k]
// VGPR0: K=0 (lanes 0-15) / K=2 (lanes 16-31); VGPR1: K=1 / K=3
// Branchless: exactly one weight is 1 per lane, others contribute exact 0.
DEV v2f atileW(const float* F, const LaneW& w) {
  float t0 = fmaf(F[0], w.l0, F[2] * w.h0);
  float t1 = fmaf(F[3], w.l1, F[5] * w.h1);
  float t2 = fmaf(F[6], w.l2, F[8] * w.h2);
  float u0 = fmaf(F[1], w.l0, F[4] * w.l1);
  v2f a;
  a.x = (t0 + t1) + t2;
  a.y = fmaf(F[7], w.l2, u0);
  return a;
}
// B-tile (4x16 f32) of row-major 3x3 B: B(k,n)=B[k*3+n]
DEV v2f btileW(const float* B, const LaneW& w) {
  float t0 = fmaf(B[0], w.l0, B[6] * w.h0);
  float t1 = fmaf(B[1], w.l1, B[7] * w.h1);
  float t2 = fmaf(B[2], w.l2, B[8] * w.h2);
  float u0 = fmaf(B[3], w.l0, B[4] * w.l1);
  v2f b;
  b.x = (t0 + t1) + t2;
  b.y = fmaf(B[5], w.l2, u0);
  return b;
}
// B-tile of transpose(F) == A-tile pattern of F
DEV v2f btileTW(const float* F, const LaneW& w) { return atileW(F, w); }
// B-tile: cols 0..2 = P (3x3), col 3 = v (3-vector)
DEV v2f btileP1W(const float* P, const float* v, const LaneW& w) {
  float t0 = fmaf(P[0], w.l0, P[6] * w.h0);
  float t1 = fmaf(P[1], w.l1, P[7] * w.h1);
  float t2 = fmaf(P[2], w.l2, P[8] * w.h2);
  float t3 = fmaf(v[0], w.l3, v[2] * w.h3);
  float u0 = fmaf(P[3], w.l0, P[4] * w.l1);
  float u1 = fmaf(P[5], w.l2, v[1] * w.l3);
  v2f b;
  b.x = (t0 + t1) + (t2 + t3);
  b.y = u0 + u1;
  return b;
}
// C-tile from row-major 3x3 (VGPR v: row M=v lanes 0-15, col = lane)
DEV v8f ctileW(const float* C, const LaneW& w) {
  v8f c = {};
  c[0] = fmaf(C[0], w.l0, fmaf(C[1], w.l1, C[2] * w.l2));
  c[1] = fmaf(C[3], w.l0, fmaf(C[4], w.l1, C[5] * w.l2));
  c[2] = fmaf(C[6], w.l0, fmaf(C[7], w.l1, C[8] * w.l2));
  return c;
}

DEV v8f wmma4(v2f a, v2f b, v8f c) {
  // (neg_a, A, neg_b, B, c_mod, C, reuse_a, reuse_b)
  return __builtin_amdgcn_wmma_f32_16x16x4_f32(false, a, false, b, (short)0, c,
                                               false, false);
}

// Re-layout D-tile (rows 0..2 valid, clean zeros elsewhere) into a B-tile:
// b.x = row0 (lo) / row2 (hi via half-wave shift); b.y = row1 (lo) / 0
DEV v2f dtob(v8f d, int m, bool lo) {
  float hi2 = __shfl((float)d[2], m, 32);
  v2f b;
  b.x = lo ? (float)d[0] : hi2;
  b.y = lo ? (float)d[1] : 0.0f;
  return b;
}

// Extract 3x3 D-tile to wave-uniform scalars via v_readlane
DEV void dget3(v8f d, float* R) {
  float r0 = d[0], r1 = d[1], r2 = d[2];
  R[0] = rdl(r0, 0); R[1] = rdl(r0, 1); R[2] = rdl(r0, 2);
  R[3] = rdl(r1, 0); R[4] = rdl(r1, 1); R[5] = rdl(r1, 2);
  R[6] = rdl(r2, 0); R[7] = rdl(r2, 1); R[8] = rdl(r2, 2);
}

__global__ __launch_bounds__(32, 1)
void EKF_63067299774705_kernel(const float* __restrict__ X,
                               const float* __restrict__ Y,
                               const float* __restrict__ Q,
                               const float* __restrict__ R,
                               float* __restrict__ out, int NT) {
  const int l = threadIdx.x;
  const int m = l & 15;
  const bool lo = l < 16;
  const float DT = 0.02f;

  float Qm[9];
#pragma unroll
  for (int i = 0; i < 9; ++i) Qm[i] = Q[i];
  const float Rd0 = R[0], Rd1 = R[4], Rd2 = R[8];

  // loop-invariant per-lane layout weights
  LaneW w;
  w.l0 = (lo && m == 0) ? 1.0f : 0.0f;
  w.l1 = (lo && m == 1) ? 1.0f : 0.0f;
  w.l2 = (lo && m == 2) ? 1.0f : 0.0f;
  w.l3 = (lo && m == 3) ? 1.0f : 0.0f;
  w.h0 = (!lo && m == 0) ? 1.0f : 0.0f;
  w.h1 = (!lo && m == 1) ? 1.0f : 0.0f;
  w.h2 = (!lo && m == 2) ? 1.0f : 0.0f;
  w.h3 = (!lo && m == 3) ? 1.0f : 0.0f;

  // loop-invariant C-tiles: Q, s*I patterns
  const v8f cQ = ctileW(Qm, w);
  v8f cI = {};  cI[0] = w.l0; cI[1] = w.l1; cI[2] = w.l2;          // 1*I
  v8f c6 = {};  c6[0] = w.l0 * (1.0f / 6.0f); c6[1] = w.l1 * (1.0f / 6.0f);
                c6[2] = w.l2 * (1.0f / 6.0f);                      // (1/6)*I
  v8f ch = {};  ch[0] = w.l0 * 0.5f; ch[1] = w.l1 * 0.5f;
                ch[2] = w.l2 * 0.5f;                               // (1/2)*I
  const v8f zc = {};

  // per-lane output select weights and addressing:
  // lanes 0-2 -> traj element l, lanes 3-11 -> Pk element l-3
  float sw[12];
#pragma unroll
  for (int j = 0; j < 12; ++j) sw[j] = (l == j) ? 1.0f : 0.0f;
  int off = (l < 3) ? l : (3 * NT + (l - 3));
  const int strd = (l < 3) ? 3 : 9;

  float xs0 = 0.0f, xs1 = 0.0f, xs2 = 0.0f;
  float Pm[9] = {1, 0, 0, 0, 1, 0, 0, 0, 1};

  float acc = 0.0f, total = 0.0f;
  int c5 = 499;  // branchless sequence-boundary counter

  for (int t = 0; t < NT; ++t) {
    const float y0 = Y[3 * t], y1 = Y[3 * t + 1], y2 = Y[3 * t + 2];
    const float xr0 = X[3 * t], xr1 = X[3 * t + 1], xr2 = X[3 * t + 2];

    // M = DELTA * Jacobian(z), z = x[0]
    const float z = xs0;
    float Mm[9];
    Mm[0] = -10.0f * DT; Mm[1] = 10.0f * DT; Mm[2] = 0.0f;
    Mm[3] = 28.0f * DT;  Mm[4] = -1.0f * DT; Mm[5] = -z * DT;
    Mm[6] = 0.0f;        Mm[7] = z * DT;     Mm[8] = (-8.0f / 3.0f) * DT;

    // F = sum_{j=0..5} M^j / j!  (Horner; chain stays in tile registers)
    v2f aM = atileW(Mm, w);
    float X4s[9];
#pragma unroll
    for (int i = 0; i < 9; ++i)
      X4s[i] = Mm[i] * (1.0f / 120.0f) +
               ((i == 0 || i == 4 || i == 8) ? (1.0f / 24.0f) : 0.0f);
    v2f b = btileW(X4s, w);
    v8f d = wmma4(aM, b, c6);
    b = dtob(d, m, lo);
    d = wmma4(aM, b, ch);
    b = dtob(d, m, lo);
    d = wmma4(aM, b, cI);
    b = dtob(d, m, lo);
    d = wmma4(aM, b, cI);
    float Fm[9];
    dget3(d, Fm);

    // [F*P | F*x] in one WMMA (x rides as 4th B column)
    float xv[3] = {xs0, xs1, xs2};
    d = wmma4(atileW(Fm, w), btileP1W(Pm, xv, w), zc);
    float FP[9];
    dget3(d, FP);
    const float xn0 = rdl(d[0], 3), xn1 = rdl(d[1], 3), xn2 = rdl(d[2], 3);

    // P_neg = (F P) F^T + Q  (Q fused into accumulator), then symmetrize
    d = wmma4(atileW(FP, w), btileTW(Fm, w), cQ);
    float Pn[9];
    dget3(d, Pn);
    { float s;
      s = 0.5f * (Pn[1] + Pn[3]); Pn[1] = s; Pn[3] = s;
      s = 0.5f * (Pn[2] + Pn[6]); Pn[2] = s; Pn[6] = s;
      s = 0.5f * (Pn[5] + Pn[7]); Pn[5] = s; Pn[7] = s; }

    // Re = P_neg + diag(R); symmetric 3x3 inverse by adjugate
    const float ra = Pn[0] + Rd0, rb = Pn[1], rc = Pn[2];
    const float rd_ = Pn[4] + Rd1, re_ = Pn[5], rf = Pn[8] + Rd2;
    const float i00 = rd_ * rf - re_ * re_;
    const float i01 = rc * re_ - rb * rf;
    const float i02 = rb * re_ - rc * rd_;
    const float i11 = ra * rf - rc * rc;
    const float i12 = rb * rc - ra * re_;
    const float i22 = ra * rd_ - rb * rb;
    const float det = ra * i00 + rb * i01 + rc * i02;
    const float dv = 1.0f / det;
    float Ri[9] = {i00 * dv, i01 * dv, i02 * dv,
                   i01 * dv, i11 * dv, i12 * dv,
                   i02 * dv, i12 * dv, i22 * dv};

    // K = P_neg * Re^{-1}
    d = wmma4(atileW(Pn, w), btileW(Ri, w), zc);
    float Km[9];
    dget3(d, Km);

    // one WMMA: [K*P_neg | K*(y - x_neg)]
    const float iv[3] = {y0 - xn0, y1 - xn1, y2 - xn2};
    d = wmma4(atileW(Km, w), btileP1W(Pn, iv, w), zc);
    float KP[9];
    dget3(d, KP);
    const float xp0 = xn0 + rdl(d[0], 3);
    const float xp1 = xn1 + rdl(d[1], 3);
    const float xp2 = xn2 + rdl(d[2], 3);

    // P_pos = P_neg - K*P_neg, symmetrized
    float Pp[9];
#pragma unroll
    for (int i = 0; i < 9; ++i) Pp[i] = Pn[i] - KP[i];
    { float s;
      s = 0.5f * (Pp[1] + Pp[3]); Pp[1] = s; Pp[3] = s;
      s = 0.5f * (Pp[2] + Pp[6]); Pp[2] = s; Pp[6] = s;
      s = 0.5f * (Pp[5] + Pp[7]); Pp[5] = s; Pp[7] = s; }

    // mse accumulation (per-500-step sequence Frobenius norms), branchless
    const float dx0 = xp0 - xr0, dx1 = xp1 - xr1, dx2 = xp2 - xr2;
    acc += dx0 * dx0 + dx1 * dx1 + dx2 * dx2;
    const bool bnd = (c5 == 0);
    total += bnd ? sqrtf(acc) : 0.0f;
    acc = bnd ? 0.0f : acc;
    c5 = bnd ? 499 : (c5 - 1);

    // outputs: traj[3t+i] (lanes 0-2), Pk[3*NT + 9t + j] (lanes 3-11)
    float sv = xp0 * sw[0];
    sv = fmaf(xp1, sw[1], sv);  sv = fmaf(xp2, sw[2], sv);
    sv = fmaf(Pp[0], sw[3], sv); sv = fmaf(Pp[1], sw[4], sv);
    sv = fmaf(Pp[2], sw[5], sv); sv = fmaf(Pp[3], sw[6], sv);
    sv = fmaf(Pp[4], sw[7], sv); sv = fmaf(Pp[5], sw[8], sv);
    sv = fmaf(Pp[6], sw[9], sv); sv = fmaf(Pp[7], sw[10], sv);
    sv = fmaf(Pp[8], sw[11], sv);
    if (l < 12) out[off] = sv;
    off += strd;

    xs0 = xp0; xs1 = xp1; xs2 = xp2;
#pragma unroll
    for (int i = 0; i < 9; ++i) Pm[i] = Pp[i];
  }

  if (l == 0) {
    const int nseq = (NT >= 500) ? (NT / 500) : 1;
    out[12 * NT] = total / (float)nseq;
  }
}

extern "C" void kernel_launch(void* const* d_in, const int* in_sizes, int n_in,
                              void* d_out, int out_size, void* d_ws,
                              size_t ws_size, hipStream_t stream) {
  (void)n_in; (void)out_size; (void)d_ws; (void)ws_size;
  const float* X = (const float*)d_in[0];
  const float* Y = (const float*)d_in[1];
  const float* Q = (const float*)d_in[2];
  const float* R = (const float*)d_in[3];
  float* out = (float*)d_out;
  const int NT = in_sizes[1] / 3;  // N*T (serial scan length)
  EKF_63067299774705_kernel<<<dim3(1), dim3(32), 0, stream>>>(X, Y, Q, R, out,
                                                              NT);
}